// ChannelIndependentConv_28913719836842
// MI455X (gfx1250) — compile-verified
//
#include <hip/hip_runtime.h>

typedef __attribute__((ext_vector_type(2))) float v2f;
typedef __attribute__((ext_vector_type(4))) float v4f;
typedef __attribute__((ext_vector_type(8))) float v8f;

// Problem constants (fixed by the reference)
#define BATCH   8
#define NNODE   256
#define IN_F    64
#define OUT_F   64
#define IN_E    32
#define OUT_E   64
#define BN      (BATCH * NNODE)   // 2048 rows for node projections

// ---------------------------------------------------------------------------
// CDNA5 async global->LDS copies (ASYNCcnt-tracked, bypass VGPRs).
// LDS address is taken from the VDST VGPR; the low 32 bits of a generic pointer into
// __shared__ are exactly the addrspace(3) offset the hardware expects.
// ---------------------------------------------------------------------------
__device__ __forceinline__ void async_copy_b128(void* lds_dst, const void* gsrc) {
    unsigned ldsoff = (unsigned)(uintptr_t)lds_dst;
    unsigned long long ga = (unsigned long long)(uintptr_t)gsrc;
    asm volatile("global_load_async_to_lds_b128 %0, %1, off"
                 :: "v"(ldsoff), "v"(ga) : "memory");
}
__device__ __forceinline__ void async_copy_b32(void* lds_dst, const void* gsrc) {
    unsigned ldsoff = (unsigned)(uintptr_t)lds_dst;
    unsigned long long ga = (unsigned long long)(uintptr_t)gsrc;
    asm volatile("global_load_async_to_lds_b32 %0, %1, off"
                 :: "v"(ldsoff), "v"(ga) : "memory");
}
__device__ __forceinline__ void async_wait0() {
    asm volatile("s_wait_asynccnt 0x0" ::: "memory");
}

// ---------------------------------------------------------------------------
// Kernel 1: node_x = emb_node @ W_node + b_node
//           sxr    = relu(emb_node @ W_nodes + b_nodes)
// 2048x64 output, K=64. One 16x16 tile per wave, two accumulators sharing A.
// 512 tiles / 8 waves = 64 blocks.
// ---------------------------------------------------------------------------
__global__ __launch_bounds__(256) void node_proj_kernel(
    const float* __restrict__ emb_node,
    const float* __restrict__ Wn, const float* __restrict__ bn,
    const float* __restrict__ Ws, const float* __restrict__ bs,
    float* __restrict__ node_x, float* __restrict__ sxr)
{
    const int wave = threadIdx.x >> 5;
    const int lane = threadIdx.x & 31;
    const int half = lane >> 4;       // 0: lanes 0-15, 1: lanes 16-31
    const int l16  = lane & 15;

    const int tile = blockIdx.x * 8 + wave;   // 0..511
    const int mt = tile >> 2;                 // 0..127 M-tile
    const int ft = tile & 3;                  // 0..3   F-tile
    const int m0 = mt * 16;
    const int col = ft * 16 + l16;

    v8f acc0 = {};
    v8f acc1 = {};
    const int arow = m0 + l16;

#pragma unroll
    for (int kk = 0; kk < 16; ++kk) {
        const int k = kk * 4 + 2 * half;  // A layout: lanes0-15 K=0,1 ; lanes16-31 K=2,3
        v2f a;
        a.x = emb_node[arow * IN_F + k];
        a.y = emb_node[arow * IN_F + k + 1];
        v2f b0, b1;
        b0.x = Wn[k * OUT_F + col];
        b0.y = Wn[(k + 1) * OUT_F + col];
        b1.x = Ws[k * OUT_F + col];
        b1.y = Ws[(k + 1) * OUT_F + col];
        acc0 = __builtin_amdgcn_wmma_f32_16x16x4_f32(false, a, false, b0,
                                                     (short)0, acc0, false, false);
        acc1 = __builtin_amdgcn_wmma_f32_16x16x4_f32(false, a, false, b1,
                                                     (short)0, acc1, false, false);
    }

    const float bias_n = bn[col];
    const float bias_s = bs[col];
#pragma unroll
    for (int g = 0; g < 8; ++g) {
        const int row = m0 + g + 8 * half;    // C/D layout: VGPR g -> M=g (+8 for hi lanes)
        node_x[row * OUT_F + col] = acc0[g] + bias_n;
        float s = acc1[g] + bias_s;
        sxr[row * OUT_F + col] = s > 0.f ? s : 0.f;
    }
}

// ---------------------------------------------------------------------------
// Kernel 2: one block per (b,n). Computes
//   edge_x[m,f]  = emb_edge[b,n,m,:] @ W_edge + b_edge        (256x64, K=32)
//   edge_out     = relu(edge_x)
//   agg[f]       = sum_m A[b,n,m] * edge_x[m,f] * node_x[b,m,f]
//   node_out[f]  = relu(agg[f]) + relu(node_sx[b,n,f])
// emb_edge slab + A row + W_edge staged in LDS via async-to-LDS; fp32 WMMA;
// LDS-atomic (ds_add_f32) aggregation.
// ---------------------------------------------------------------------------
__global__ __launch_bounds__(256) void edge_fused_kernel(
    const float* __restrict__ Aadj,       // [B,N,N]
    const float* __restrict__ emb_edge,   // [B,N,N,IN_E]
    const float* __restrict__ We,         // [IN_E, OUT_E]
    const float* __restrict__ be,         // [OUT_E]
    const float* __restrict__ node_x,     // ws [B,N,OUT_F]
    const float* __restrict__ sxr,        // ws [B,N,OUT_F] (already ReLU'd)
    float* __restrict__ node_out,         // [B,N,OUT_F]
    float* __restrict__ edge_out)         // [B,N,N,OUT_E]
{
    __shared__ float sE[NNODE * IN_E];    // 32 KB: emb_edge[b,n,:,:]
    __shared__ float sW[IN_E * OUT_E];    // 8 KB : W_edge
    __shared__ float sA[NNODE];           // 1 KB : A[b,n,:]
    __shared__ float sAgg[OUT_E];

    const int bid = blockIdx.x;           // b*N + n
    const int b   = bid >> 8;
    const int tid = threadIdx.x;

    // Async staging: memory -> LDS without touching VGPRs (ASYNCcnt-tracked).
    const float* eslab = emb_edge + (size_t)bid * (NNODE * IN_E);
#pragma unroll
    for (int i = 0; i < (NNODE * IN_E) / (4 * 256); ++i) {
        const int e = (i * 256 + tid) * 4;
        async_copy_b128(&sE[e], &eslab[e]);
    }
#pragma unroll
    for (int i = 0; i < (IN_E * OUT_E) / (4 * 256); ++i) {
        const int e = (i * 256 + tid) * 4;
        async_copy_b128(&sW[e], &We[e]);
    }
    async_copy_b32(&sA[tid], &Aadj[(size_t)bid * NNODE + tid]);
    if (tid < OUT_E) sAgg[tid] = 0.f;
    async_wait0();
    __syncthreads();

    const int wave = tid >> 5;
    const int lane = tid & 31;
    const int half = lane >> 4;
    const int l16  = lane & 15;

    float* eo = edge_out + (size_t)bid * (NNODE * OUT_E);
    const float* nxb = node_x + (size_t)b * (NNODE * OUT_F);

    // 16 M-tiles (m=256/16), 2 per wave; A fragments reused over 4 F-tiles
    for (int mt = wave; mt < 16; mt += 8) {
        const int m0 = mt * 16;
        const int arow = m0 + l16;
        v2f afr[8];
#pragma unroll
        for (int kk = 0; kk < 8; ++kk) {
            const int k = kk * 4 + 2 * half;
            afr[kk].x = sE[arow * IN_E + k];
            afr[kk].y = sE[arow * IN_E + k + 1];
        }
#pragma unroll
        for (int ft = 0; ft < 4; ++ft) {
            const int col = ft * 16 + l16;
            v8f acc = {};
#pragma unroll
            for (int kk = 0; kk < 8; ++kk) {
                const int k = kk * 4 + 2 * half;
                v2f bf;
                bf.x = sW[k * OUT_E + col];
                bf.y = sW[(k + 1) * OUT_E + col];
                acc = __builtin_amdgcn_wmma_f32_16x16x4_f32(false, afr[kk], false, bf,
                                                            (short)0, acc, false, false);
            }
            const float bias = be[col];
            float part = 0.f;
#pragma unroll
            for (int g = 0; g < 8; ++g) {
                const int m = m0 + g + 8 * half;
                const float ex = acc[g] + bias;         // edge_x (pre-ReLU, with bias)
                eo[m * OUT_E + col] = ex > 0.f ? ex : 0.f;
                part += sA[m] * ex * nxb[m * OUT_F + col];
            }
            atomicAdd(&sAgg[col], part);                // ds_add_f32
        }
    }
    __syncthreads();

    if (tid < OUT_F) {
        float v = sAgg[tid];
        v = v > 0.f ? v : 0.f;
        node_out[(size_t)bid * OUT_F + tid] = v + sxr[(size_t)bid * OUT_F + tid];
    }
}

// ---------------------------------------------------------------------------
// Launch. Inputs (setup_inputs order):
//   0:A  1:emb_node  2:emb_edge  3:W_node  4:b_node  5:W_nodes  6:b_nodes
//   7:W_edge  8:b_edge
// Output: node_out [8,256,64] then edge_out [8,256,256,64], concatenated.
// Workspace: node_x (512 KB) + relu(node_sx) (512 KB).
// ---------------------------------------------------------------------------
extern "C" void kernel_launch(void* const* d_in, const int* in_sizes, int n_in,
                              void* d_out, int out_size, void* d_ws, size_t ws_size,
                              hipStream_t stream) {
    const float* Aadj     = (const float*)d_in[0];
    const float* emb_node = (const float*)d_in[1];
    const float* emb_edge = (const float*)d_in[2];
    const float* W_node   = (const float*)d_in[3];
    const float* b_node   = (const float*)d_in[4];
    const float* W_nodes  = (const float*)d_in[5];
    const float* b_nodes  = (const float*)d_in[6];
    const float* W_edge   = (const float*)d_in[7];
    const float* b_edge   = (const float*)d_in[8];

    float* node_out = (float*)d_out;                       // 8*256*64
    float* edge_out = (float*)d_out + BN * OUT_F;          // 8*256*256*64

    float* node_x = (float*)d_ws;                          // BN*64 floats
    float* sxr    = (float*)d_ws + BN * OUT_F;             // BN*64 floats

    node_proj_kernel<<<64, 256, 0, stream>>>(emb_node, W_node, b_node,
                                             W_nodes, b_nodes, node_x, sxr);
    edge_fused_kernel<<<BN, 256, 0, stream>>>(Aadj, emb_edge, W_edge, b_edge,
                                              node_x, sxr, node_out, edge_out);
}